// SelfAttention3D_14156212208375
// MI455X (gfx1250) — compile-verified
//
#include <hip/hip_runtime.h>
#include <math.h>

// Problem constants (from reference): visual [B,C,N], W_channel [H,N], W_spatial [H,C]
#define B_ 32
#define C_ 512
#define N_ 2048
#define H_ 512

typedef __attribute__((ext_vector_type(2))) float v2f;
typedef __attribute__((ext_vector_type(8))) float v8f;

// ---------------------------------------------------------------------------
// 1) vsum[b,n] = sum_c visual[b,c,n]   (coalesced over n, 8KB-stride walk in c)
// ---------------------------------------------------------------------------
__global__ void k_colsum(const float* __restrict__ vis, float* __restrict__ vsum) {
    const int n = blockIdx.x * 256 + threadIdx.x;
    const int b = blockIdx.y;
    const float* p = vis + (size_t)b * C_ * N_ + n;
    float acc = 0.f;
    #pragma unroll 8
    for (int c = 0; c < C_; ++c) {
        if (c + 16 < C_) __builtin_prefetch(&p[(size_t)(c + 16) * N_], 0, 0);
        acc += p[(size_t)c * N_];
    }
    vsum[b * N_ + n] = acc;
}

// ---------------------------------------------------------------------------
// 2) rowdot: out[b,c] = sum_n visual[b,c,n] * (USEVEC ? vec[b,n] : 1)
//    one block per (b,c) row; LDS tree reduction
// ---------------------------------------------------------------------------
template <bool USEVEC>
__global__ void k_rowdot(const float* __restrict__ vis, const float* __restrict__ vec,
                         float* __restrict__ out) {
    const int bc = blockIdx.x;
    const int b  = bc / C_;
    const float* p = vis + (size_t)bc * N_;
    const float* v = vec + (size_t)b * N_;
    float acc = 0.f;
    for (int n = threadIdx.x; n < N_; n += 256)
        acc += USEVEC ? (p[n] * v[n]) : p[n];
    __shared__ float s[256];
    s[threadIdx.x] = acc;
    __syncthreads();
    for (int off = 128; off > 0; off >>= 1) {
        if (threadIdx.x < off) s[threadIdx.x] += s[threadIdx.x + off];
        __syncthreads();
    }
    if (threadIdx.x == 0) out[bc] = s[0];
}

// ---------------------------------------------------------------------------
// 3) WMMA fp32 GEMM: D[32 x Nout] = A[32 x K] * B, B(k,n) = Bm[k*ldbk + n*ldbn]
//    One 16x16 tile per wave32 via V_WMMA_F32_16X16X4_F32.
//    A-frag layout: lane = m + 16*(k>>1), vgpr = k&1  (ISA 7.12.2)
//    B-frag layout: mirrored with n in place of m
//    D layout: vgpr r -> M = 8*(lane>>4) + r, N = lane&15
//    Grid sized so every wave owns a tile -> EXEC all ones (WMMA requirement).
// ---------------------------------------------------------------------------
__global__ void k_wmma_gemm(const float* __restrict__ A, const float* __restrict__ Bm,
                            float* __restrict__ D, int K, int Nout, int ldbk, int ldbn) {
    const int wave = (blockIdx.x * 256 + threadIdx.x) >> 5;
    const int lane = threadIdx.x & 31;
    const int tM = wave & 1;        // M = 32 -> 2 row tiles
    const int tN = wave >> 1;
    const int mn   = lane & 15;     // row index for A-frag, col index for B-frag
    const int half = lane >> 4;     // selects K pair {0,1} vs {2,3}
    const float* Arow = A  + (size_t)(tM * 16 + mn) * K;
    const float* Bcol = Bm + (size_t)(tN * 16 + mn) * ldbn;
    v8f acc = {};
    for (int k0 = 0; k0 < K; k0 += 4) {
        const int ks = k0 + 2 * half;
        v2f a, b;
        a.x = Arow[ks];
        a.y = Arow[ks + 1];
        b.x = Bcol[(size_t)ks * ldbk];
        b.y = Bcol[(size_t)(ks + 1) * ldbk];
        acc = __builtin_amdgcn_wmma_f32_16x16x4_f32(
            /*neg_a=*/false, a, /*neg_b=*/false, b,
            /*c_mod=*/(short)0, acc, /*reuse_a=*/false, /*reuse_b=*/false);
    }
    const int nOut  = tN * 16 + mn;
    const int mBase = tM * 16 + half * 8;
    #pragma unroll
    for (int r = 0; r < 8; ++r)
        D[(size_t)(mBase + r) * Nout + nOut] = acc[r];
}

// ---------------------------------------------------------------------------
// 4) channel softmax over C per batch; fused fr[b,c] = mask[b,c] * vrow[b,c]
// ---------------------------------------------------------------------------
__global__ void k_softmax_c(const float* __restrict__ sim, const float* __restrict__ vrow,
                            float* __restrict__ cmask, float* __restrict__ fr) {
    const int b = blockIdx.x;
    const float* x = sim + b * C_;
    __shared__ float s[256];
    float m = -INFINITY;
    for (int i = threadIdx.x; i < C_; i += 256) m = fmaxf(m, x[i]);
    s[threadIdx.x] = m;
    __syncthreads();
    for (int off = 128; off > 0; off >>= 1) {
        if (threadIdx.x < off) s[threadIdx.x] = fmaxf(s[threadIdx.x], s[threadIdx.x + off]);
        __syncthreads();
    }
    const float mx = s[0];
    __syncthreads();
    float sum = 0.f;
    for (int i = threadIdx.x; i < C_; i += 256) sum += expf(x[i] - mx);
    s[threadIdx.x] = sum;
    __syncthreads();
    for (int off = 128; off > 0; off >>= 1) {
        if (threadIdx.x < off) s[threadIdx.x] += s[threadIdx.x + off];
        __syncthreads();
    }
    const float inv = 1.0f / s[0];
    for (int i = threadIdx.x; i < C_; i += 256) {
        const float v = expf(x[i] - mx) * inv;
        cmask[b * C_ + i] = v;
        fr[b * C_ + i]    = v * vrow[b * C_ + i];
    }
}

// ---------------------------------------------------------------------------
// 5) sims[b,n] = sum_c alpha[b,c] * visual[b,c,n], alpha = wsg*mask (LDS-cached)
// ---------------------------------------------------------------------------
__global__ void k_colsum_weighted(const float* __restrict__ vis, const float* __restrict__ wsg,
                                  const float* __restrict__ cm, float* __restrict__ sims) {
    __shared__ float alpha[C_];
    const int b = blockIdx.y;
    for (int c = threadIdx.x; c < C_; c += 256)
        alpha[c] = wsg[b * C_ + c] * cm[b * C_ + c];
    __syncthreads();
    const int n = blockIdx.x * 256 + threadIdx.x;
    const float* p = vis + (size_t)b * C_ * N_ + n;
    float acc = 0.f;
    #pragma unroll 8
    for (int c = 0; c < C_; ++c) {
        if (c + 16 < C_) __builtin_prefetch(&p[(size_t)(c + 16) * N_], 0, 0);
        acc += alpha[c] * p[(size_t)c * N_];
    }
    sims[b * N_ + n] = acc;
}

// ---------------------------------------------------------------------------
// 6) spatial softmax over N per batch
// ---------------------------------------------------------------------------
__global__ void k_softmax_s(const float* __restrict__ sim, float* __restrict__ smask) {
    const int b = blockIdx.x;
    const float* x = sim + b * N_;
    __shared__ float s[256];
    float m = -INFINITY;
    for (int i = threadIdx.x; i < N_; i += 256) m = fmaxf(m, x[i]);
    s[threadIdx.x] = m;
    __syncthreads();
    for (int off = 128; off > 0; off >>= 1) {
        if (threadIdx.x < off) s[threadIdx.x] = fmaxf(s[threadIdx.x], s[threadIdx.x + off]);
        __syncthreads();
    }
    const float mx = s[0];
    __syncthreads();
    float sum = 0.f;
    for (int i = threadIdx.x; i < N_; i += 256) sum += expf(x[i] - mx);
    s[threadIdx.x] = sum;
    __syncthreads();
    for (int off = 128; off > 0; off >>= 1) {
        if (threadIdx.x < off) s[threadIdx.x] += s[threadIdx.x + off];
        __syncthreads();
    }
    const float inv = 1.0f / s[0];
    for (int i = threadIdx.x; i < N_; i += 256)
        smask[b * N_ + i] = expf(x[i] - mx) * inv;
}

// ---------------------------------------------------------------------------
// 7) out[b,c,n] = visual[b,c,n] * cmask[b,c] * smask[b,n]  (float4 streams)
// ---------------------------------------------------------------------------
__global__ void k_final(const float4* __restrict__ vis4, const float* __restrict__ cm,
                        const float* __restrict__ sm, float4* __restrict__ out4) {
    const size_t i4   = (size_t)blockIdx.x * 256 + threadIdx.x;
    const size_t base = i4 * 4;
    const int    n = (int)(base % N_);
    const size_t t = base / N_;
    const int    c = (int)(t % C_);
    const int    b = (int)(t / C_);
    const float  cmv = cm[b * C_ + c];
    const float4 v = vis4[i4];
    const float4 s = *(const float4*)(sm + (size_t)b * N_ + n);
    float4 o;
    o.x = v.x * cmv * s.x;
    o.y = v.y * cmv * s.y;
    o.z = v.z * cmv * s.z;
    o.w = v.w * cmv * s.w;
    out4[i4] = o;
}

// ---------------------------------------------------------------------------
extern "C" void kernel_launch(void* const* d_in, const int* in_sizes, int n_in,
                              void* d_out, int out_size, void* d_ws, size_t ws_size,
                              hipStream_t stream) {
    (void)in_sizes; (void)n_in; (void)out_size; (void)ws_size;
    const float* vis = (const float*)d_in[0];
    const float* Wcf = (const float*)d_in[1];  // [H, N]
    const float* Wcg = (const float*)d_in[2];  // [H, N]
    const float* Wsf = (const float*)d_in[3];  // [H, C]
    const float* Wsg = (const float*)d_in[4];  // [H, C]

    float* out       = (float*)d_out;                    // [B,C,N]
    float* cmask_out = out + (size_t)B_ * C_ * N_;       // [B,C]
    float* smask_out = cmask_out + (size_t)B_ * C_;      // [B,N]

    float* ws   = (float*)d_ws;
    float* vsum = ws;  ws += B_ * N_;   // sum_c visual
    float* vrow = ws;  ws += B_ * C_;   // sum_n visual
    float* fsum = ws;  ws += B_ * H_;   // vsum * Wcf^T
    float* wgv  = ws;  ws += B_ * N_;   // fsum * Wcg
    float* simc = ws;  ws += B_ * C_;   // channel logits
    float* fr   = ws;  ws += B_ * C_;   // mask * vrow
    float* gv   = ws;  ws += B_ * H_;   // fr * Wsf^T
    float* wsgv = ws;  ws += B_ * C_;   // gv * Wsg
    float* sims = ws;  ws += B_ * N_;   // spatial logits

    // Channel attention (collapsed algebra; no [B,C,H] tensor ever built)
    k_colsum<<<dim3(N_ / 256, B_), 256, 0, stream>>>(vis, vsum);
    k_rowdot<false><<<B_ * C_, 256, 0, stream>>>(vis, vsum /*unused*/, vrow);
    // fsum[b,h] = sum_n vsum[b,n]*Wcf[h,n] : B(k=n, col=h) -> ldbk=1, ldbn=N
    k_wmma_gemm<<<(2 * (H_ / 16)) / 8, 256, 0, stream>>>(vsum, Wcf, fsum, N_, H_, 1, N_);
    // wgv[b,n] = sum_h fsum[b,h]*Wcg[h,n] : B(k=h, col=n) -> ldbk=N, ldbn=1
    k_wmma_gemm<<<(2 * (N_ / 16)) / 8, 256, 0, stream>>>(fsum, Wcg, wgv, H_, N_, N_, 1);
    k_rowdot<true><<<B_ * C_, 256, 0, stream>>>(vis, wgv, simc);
    k_softmax_c<<<B_, 256, 0, stream>>>(simc, vrow, cmask_out, fr);

    // Spatial attention (collapsed algebra; no [B,H,N] tensor ever built)
    // gv[b,h] = sum_c fr[b,c]*Wsf[h,c] : ldbk=1, ldbn=C
    k_wmma_gemm<<<(2 * (H_ / 16)) / 8, 256, 0, stream>>>(fr, Wsf, gv, C_, H_, 1, C_);
    // wsgv[b,c] = sum_h gv[b,h]*Wsg[h,c] : ldbk=C, ldbn=1
    k_wmma_gemm<<<(2 * (C_ / 16)) / 8, 256, 0, stream>>>(gv, Wsg, wsgv, H_, C_, C_, 1);
    k_colsum_weighted<<<dim3(N_ / 256, B_), 256, 0, stream>>>(vis, wsgv, cmask_out, sims);
    k_softmax_s<<<B_, 256, 0, stream>>>(sims, smask_out);

    // Fused final elementwise: out = visual * channel_mask * spatial_mask
    k_final<<<(int)(((size_t)B_ * C_ * N_ / 4) / 256), 256, 0, stream>>>(
        (const float4*)vis, cmask_out, smask_out, (float4*)d_out);
}